// GraphSAGEEncoder_46351287058738
// MI455X (gfx1250) — compile-verified
//
#include <hip/hip_runtime.h>
#include <hip/hip_bf16.h>

typedef __attribute__((ext_vector_type(2))) float v2f;
typedef __attribute__((ext_vector_type(8))) float v8f;

#define D 64  // D_IN == D_HID == 64

// ---------------------------------------------------------------- zero
__global__ void sage_zero_kernel(float* __restrict__ p, int n) {
  int i = blockIdx.x * blockDim.x + threadIdx.x;
  if (i < n) p[i] = 0.0f;
}

// ---------------------------------------------------------------- edge scatter
// 16 threads per edge; each thread moves 4 consecutive features (float4 read,
// 4x global_atomic_add_f32). x rows are 256B -> one edge's 16 threads read a
// fully coalesced row; atomics land on consecutive addresses of the dst row.
template <bool COUNT>
__global__ __launch_bounds__(256) void sage_scatter_kernel(
    const float* __restrict__ x, const int* __restrict__ src,
    const int* __restrict__ dst, float* __restrict__ summed,
    float* __restrict__ cnt, int nE) {
  int tid = blockIdx.x * blockDim.x + threadIdx.x;
  int e = tid >> 4;
  int c = tid & 15;
  if (e >= nE) return;
  int s = src[e];
  int d = dst[e];
  const float4 v = *(const float4*)(x + (size_t)s * D + c * 4);
  float* o = summed + (size_t)d * D + c * 4;
  atomicAdd(o + 0, v.x);
  atomicAdd(o + 1, v.y);
  atomicAdd(o + 2, v.z);
  atomicAdd(o + 3, v.w);
  if (COUNT && c == 0) atomicAdd(cnt + d, 1.0f);
}

// ---------------------------------------------------------------- WMMA GEMM
// out[i,:] = (summed[i,:]/max(cnt[i],1)) @ Wl^T + xin[i,:] @ Wr^T + b  (opt ReLU)
// One wave -> 16 rows x 64 cols (4 accumulator tiles). 8 waves/block -> 128 rows.
// Full-precision path via V_WMMA_F32_16X16X4_F32 (K stepped by 4).
__global__ __launch_bounds__(256) void sage_gemm_kernel(
    const float* __restrict__ summed, const float* __restrict__ xin,
    const float* __restrict__ cnt, const float* __restrict__ Wl,
    const float* __restrict__ Wr, const float* __restrict__ bias,
    float* __restrict__ out, int nNodes, int relu) {
  const int lane = threadIdx.x & 31;
  const int wave = threadIdx.x >> 5;
  const int half = lane >> 4;   // selects K pair within a 4-wide K step
  const int ln16 = lane & 15;   // A: row M ; B: col N
  const long rowBase = (long)blockIdx.x * 128 + (long)wave * 16;
  long aRow = rowBase + ln16;
  if (aRow >= nNodes) aRow = nNodes - 1;  // clamp loads: keep EXEC all-ones for WMMA

  const float invc = 1.0f / fmaxf(cnt[aRow], 1.0f);
  const float* __restrict__ aggRow = summed + (size_t)aRow * D;
  const float* __restrict__ xRow = xin + (size_t)aRow * D;

  v8f acc[4] = {v8f{}, v8f{}, v8f{}, v8f{}};

  // ---- phase 1: mean-aggregated neighbors through Wl^T
  for (int kk = 0; kk < 16; ++kk) {
    const int k0 = kk * 4 + half * 2;
    v2f a;
    a.x = aggRow[k0] * invc;
    a.y = aggRow[k0 + 1] * invc;
#pragma unroll
    for (int nt = 0; nt < 4; ++nt) {
      const float* __restrict__ w = Wl + (size_t)(nt * 16 + ln16) * D + k0;
      v2f b;
      b.x = w[0];
      b.y = w[1];
      acc[nt] = __builtin_amdgcn_wmma_f32_16x16x4_f32(
          false, a, false, b, (short)0, acc[nt], false, false);
    }
  }
  // ---- phase 2: self features through Wr^T
  for (int kk = 0; kk < 16; ++kk) {
    const int k0 = kk * 4 + half * 2;
    v2f a;
    a.x = xRow[k0];
    a.y = xRow[k0 + 1];
#pragma unroll
    for (int nt = 0; nt < 4; ++nt) {
      const float* __restrict__ w = Wr + (size_t)(nt * 16 + ln16) * D + k0;
      v2f b;
      b.x = w[0];
      b.y = w[1];
      acc[nt] = __builtin_amdgcn_wmma_f32_16x16x4_f32(
          false, a, false, b, (short)0, acc[nt], false, false);
    }
  }

  // ---- epilogue: bias, ReLU, guarded store
  // C/D layout: element v of lane -> M = v + 8*half, N = ln16
#pragma unroll
  for (int nt = 0; nt < 4; ++nt) {
    const float bv = bias[nt * 16 + ln16];
#pragma unroll
    for (int v = 0; v < 8; ++v) {
      const long m = rowBase + v + 8 * half;
      if (m < nNodes) {
        float val = acc[nt][v] + bv;
        if (relu) val = fmaxf(val, 0.0f);
        out[m * D + nt * 16 + ln16] = val;
      }
    }
  }
}

// ---------------------------------------------------------------- launch
extern "C" void kernel_launch(void* const* d_in, const int* in_sizes, int n_in,
                              void* d_out, int out_size, void* d_ws,
                              size_t ws_size, hipStream_t stream) {
  const float* x = (const float*)d_in[0];
  const int* edge = (const int*)d_in[1];
  const float* W1l = (const float*)d_in[2];
  const float* W1r = (const float*)d_in[3];
  const float* b1 = (const float*)d_in[4];
  const float* W2l = (const float*)d_in[5];
  const float* W2r = (const float*)d_in[6];
  const float* b2 = (const float*)d_in[7];

  const int N = in_sizes[0] / D;   // 100000
  const int E = in_sizes[1] / 2;   // 1200000
  const int* src = edge;
  const int* dst = edge + E;

  // workspace layout: h1 [N*D] | summed [N*D] | cnt [N]   (contiguous)
  float* h1 = (float*)d_ws;
  float* summed = h1 + (size_t)N * D;
  float* cntb = summed + (size_t)N * D;
  float* outp = (float*)d_out;

  const int scatterBlocks = (E * 16 + 255) / 256;
  const int gemmBlocks = (N + 127) / 128;

  // ---- layer 1
  {
    int zn = N * (D + 1);  // summed and cnt are contiguous -> one zero pass
    sage_zero_kernel<<<(zn + 255) / 256, 256, 0, stream>>>(summed, zn);
  }
  sage_scatter_kernel<true>
      <<<scatterBlocks, 256, 0, stream>>>(x, src, dst, summed, cntb, E);
  sage_gemm_kernel<<<gemmBlocks, 256, 0, stream>>>(summed, x, cntb, W1l, W1r,
                                                   b1, h1, N, /*relu=*/1);

  // ---- layer 2 (degrees unchanged: keep cnt, re-zero summed only)
  sage_zero_kernel<<<(N * D + 255) / 256, 256, 0, stream>>>(summed, N * D);
  sage_scatter_kernel<false>
      <<<scatterBlocks, 256, 0, stream>>>(h1, src, dst, summed, cntb, E);
  sage_gemm_kernel<<<gemmBlocks, 256, 0, stream>>>(summed, h1, cntb, W2l, W2r,
                                                   b2, outp, N, /*relu=*/0);
}